// OldAttentionBlock_7069516169835
// MI455X (gfx1250) — compile-verified
//
#include <hip/hip_runtime.h>

// Fused prior+active attention (flash-style online softmax), fp32 WMMA.
// B*H workgroups; each owns one (b,h): Q[16,256] vs KV[4096,256] + [16,256].
// Memory-bound: ~4.3 GB of K/V streamed once => ~185us floor at 23.3 TB/s.
// K/V staged to LDS via CDNA5 async-to-LDS (ASYNCcnt), triple-buffered with
// prefetch depth 2 and one barrier per tile.

#define D        256
#define LQ       16
#define LP       4096
#define NTILES   (LP / 16)
#define KSTRIDE  260   // padded LDS row stride (floats) for Q/K/V tiles
#define SSTRIDE  20    // padded LDS row stride for per-wave P tile
#define SCALE    0.0625f  // 256^-0.5

#if __has_builtin(__builtin_amdgcn_global_load_async_to_lds_b128) && \
    __has_builtin(__builtin_amdgcn_s_wait_asynccnt)
#define USE_ASYNC 1
#define NBUF 3
#else
#define USE_ASYNC 0
#define NBUF 2
#endif

typedef float v2f __attribute__((ext_vector_type(2)));
typedef float v8f __attribute__((ext_vector_type(8)));

#if USE_ASYNC
typedef int v4i __attribute__((ext_vector_type(4)));
typedef __attribute__((address_space(1))) v4i g_v4i;   // global int4
typedef __attribute__((address_space(3))) v4i l_v4i;   // LDS int4
__device__ __forceinline__ void cp_async16(const float* g, float* l) {
  __builtin_amdgcn_global_load_async_to_lds_b128((g_v4i*)g, (l_v4i*)l, 0, 0);
}
#endif

__device__ __forceinline__ float rowmax16(float x) {
  x = fmaxf(x, __shfl_xor(x, 1, 16));
  x = fmaxf(x, __shfl_xor(x, 2, 16));
  x = fmaxf(x, __shfl_xor(x, 4, 16));
  x = fmaxf(x, __shfl_xor(x, 8, 16));
  return x;
}
__device__ __forceinline__ float rowsum16(float x) {
  x += __shfl_xor(x, 1, 16);
  x += __shfl_xor(x, 2, 16);
  x += __shfl_xor(x, 4, 16);
  x += __shfl_xor(x, 8, 16);
  return x;
}

extern "C" __global__ __launch_bounds__(256)
void fa_kernel(const float* __restrict__ Q,
               const float* __restrict__ Kp,
               const float* __restrict__ Vp,
               const float* __restrict__ Ka,
               const float* __restrict__ Va,
               const unsigned char* __restrict__ Mask,  // jnp.bool_ -> 1 byte
               float* __restrict__ Out)
{
  __shared__ float ldsQ[LQ][KSTRIDE];
  __shared__ float ldsK[NBUF][LQ][KSTRIDE];
  __shared__ float ldsV[NBUF][LQ][KSTRIDE];
  __shared__ float ldsS[8][LQ][SSTRIDE];

  const int bh   = blockIdx.x;
  const int tid  = threadIdx.x;
  const int wave = tid >> 5;
  const int lane = tid & 31;
  const int half = lane >> 4;   // which 16-lane group
  const int l16  = lane & 15;

  const float*         qb = Q  + (size_t)bh * LQ * D;
  const float*         kb = Kp + (size_t)bh * LP * D;
  const float*         vb = Vp + (size_t)bh * LP * D;
  const float*         ka = Ka + (size_t)bh * LQ * D;
  const float*         va = Va + (size_t)bh * LQ * D;
  const unsigned char* mb = Mask + (size_t)bh * LQ * LP;

  // ---- stage Q tile (16x256) into LDS: 1024 float4, 4 per thread ----
  #pragma unroll
  for (int j = 0; j < 4; ++j) {
    int idx = tid + 256 * j;
    float4 v = ((const float4*)qb)[idx];
    *(float4*)&ldsQ[idx >> 6][(idx & 63) * 4] = v;
  }

  // Flash state: stats slot i maps to query row (i + 8*half) per C/D layout.
  float m_run[8], l_run[8];
  v8f o0 = {}, o1 = {};
  #pragma unroll
  for (int i = 0; i < 8; ++i) { m_run[i] = -INFINITY; l_run[i] = 0.f; }

  const int n0 = wave * 32;  // this wave's output column range [n0, n0+32)

  auto compute_tile = [&](int buf, int t, bool use_mask) {
    // ---- S = Q * K^T : 64 x v_wmma_f32_16x16x4_f32, two chains ----
    v8f c0 = {}, c1 = {};
    #pragma unroll
    for (int k0 = 0; k0 < D; k0 += 8) {
      int qc = k0 + 2 * half;  // A/B layout: lane=row(M or N), half selects K pair
      v2f a0 = *(const v2f*)&ldsQ[l16][qc];
      v2f b0 = *(const v2f*)&ldsK[buf][l16][qc];
      v2f a1 = *(const v2f*)&ldsQ[l16][qc + 4];
      v2f b1 = *(const v2f*)&ldsK[buf][l16][qc + 4];
      c0 = __builtin_amdgcn_wmma_f32_16x16x4_f32(false, a0, false, b0, (short)0, c0, false, false);
      c1 = __builtin_amdgcn_wmma_f32_16x16x4_f32(false, a1, false, b1, (short)0, c1, false, false);
    }
    v8f s;
    #pragma unroll
    for (int i = 0; i < 8; ++i) s[i] = (c0[i] + c1[i]) * SCALE;

    if (use_mask) {
      #pragma unroll
      for (int i = 0; i < 8; ++i) {
        int row = i + half * 8;
        if (!mb[(size_t)row * LP + t * 16 + l16]) s[i] = -3.4028235e38f;
      }
    }

    // ---- online softmax update + stash P in per-wave LDS tile ----
    #pragma unroll
    for (int i = 0; i < 8; ++i) {
      float mt = rowmax16(s[i]);
      float mn = fmaxf(m_run[i], mt);
      float al = __expf(m_run[i] - mn);
      float p  = __expf(s[i] - mn);
      float rs = rowsum16(p);
      l_run[i] = l_run[i] * al + rs;
      m_run[i] = mn;
      o0[i] *= al;
      o1[i] *= al;
      ldsS[wave][i + half * 8][l16] = p;   // C/D layout -> row-major P tile
    }

    // ---- O += P * V : 8 x wmma (2 col tiles x K=16 in chunks of 4) ----
    #pragma unroll
    for (int kk = 0; kk < 16; kk += 4) {
      int r = kk + 2 * half;
      v2f a  = *(const v2f*)&ldsS[wave][l16][r];  // A frag: lane=M row, half picks K pair
      v2f bA = { ldsV[buf][r][n0 + l16],      ldsV[buf][r + 1][n0 + l16] };
      v2f bB = { ldsV[buf][r][n0 + 16 + l16], ldsV[buf][r + 1][n0 + 16 + l16] };
      o0 = __builtin_amdgcn_wmma_f32_16x16x4_f32(false, a, false, bA, (short)0, o0, false, false);
      o1 = __builtin_amdgcn_wmma_f32_16x16x4_f32(false, a, false, bB, (short)0, o1, false, false);
    }
  };

#if USE_ASYNC
  // ---- async-to-LDS staging: triple buffer, prefetch depth 2 ----
  auto issue_tile = [&](const float* ksrc, const float* vsrc, int row0, int buf) {
    #pragma unroll
    for (int j = 0; j < 4; ++j) {
      int idx = tid + 256 * j;          // float4 index within 16x256 tile
      int r = idx >> 6, c = (idx & 63) * 4;
      cp_async16(ksrc + (size_t)row0 * D + (size_t)idx * 4, &ldsK[buf][r][c]);
      cp_async16(vsrc + (size_t)row0 * D + (size_t)idx * 4, &ldsV[buf][r][c]);
    }
  };

  issue_tile(kb, vb, 0, 0);
  issue_tile(kb, vb, 16, 1);

  for (int t = 0; t < NTILES; ++t) {
    __builtin_amdgcn_s_wait_asynccnt(8);   // tile t landed (in-order completion)
    __syncthreads();                       // visible to all waves
    if (t + 2 < NTILES)       issue_tile(kb, vb, (t + 2) * 16, (t + 2) % 3);
    else if (t + 2 == NTILES) issue_tile(ka, va, 0, (t + 2) % 3);  // active block
    compute_tile(t % 3, t, true);
  }

  __builtin_amdgcn_s_wait_asynccnt(0);
  __syncthreads();
  compute_tile(NTILES % 3, 0, false);      // active block (unmasked)

#else
  // ---- fallback: global->VGPR->LDS, double buffered ----
  float4 kreg[4], vreg[4];
  auto load_tile = [&](const float* ksrc, const float* vsrc, int row0) {
    #pragma unroll
    for (int j = 0; j < 4; ++j) {
      int idx = tid + 256 * j;
      kreg[j] = ((const float4*)(ksrc + (size_t)row0 * D))[idx];
      vreg[j] = ((const float4*)(vsrc + (size_t)row0 * D))[idx];
    }
  };
  auto store_tile = [&](int buf) {
    #pragma unroll
    for (int j = 0; j < 4; ++j) {
      int idx = tid + 256 * j;
      int r = idx >> 6, c = (idx & 63) * 4;
      *(float4*)&ldsK[buf][r][c] = kreg[j];
      *(float4*)&ldsV[buf][r][c] = vreg[j];
    }
  };

  load_tile(kb, vb, 0);
  store_tile(0);
  __syncthreads();
  for (int t = 0; t < NTILES; ++t) {
    int cur = t & 1;
    if (t + 1 < NTILES) load_tile(kb, vb, (t + 1) * 16);
    compute_tile(cur, t, true);
    __syncthreads();
    if (t + 1 < NTILES) store_tile(1 - cur);
    __syncthreads();
  }
  load_tile(ka, va, 0);
  store_tile(0);
  __syncthreads();
  compute_tile(0, 0, false);
#endif

  // ---- epilogue: divide by shared denominator, write out ----
  float* ob = Out + (size_t)bh * LQ * D;
  #pragma unroll
  for (int i = 0; i < 8; ++i) {
    int row = i + half * 8;
    float inv = 1.0f / l_run[i];
    ob[row * D + n0 + l16]      = o0[i] * inv;
    ob[row * D + n0 + 16 + l16] = o1[i] * inv;
  }
}

extern "C" void kernel_launch(void* const* d_in, const int* in_sizes, int n_in,
                              void* d_out, int out_size, void* d_ws, size_t ws_size,
                              hipStream_t stream) {
  const float* Q  = (const float*)d_in[0];
  const float* Kp = (const float*)d_in[1];
  const float* Vp = (const float*)d_in[2];
  const float* Ka = (const float*)d_in[3];
  const float* Va = (const float*)d_in[4];
  const unsigned char* M = (const unsigned char*)d_in[5];
  float* Out = (float*)d_out;

  const int BH = in_sizes[0] / (LQ * D);  // = 256 for B=8,H=32
  hipLaunchKernelGGL(fa_kernel, dim3(BH), dim3(256), 0, stream,
                     Q, Kp, Vp, Ka, Va, M, Out);
}